// RasterSSM2D_31593779429651
// MI455X (gfx1250) — compile-verified
//
#include <hip/hip_runtime.h>
#include <math.h>

// RasterSSM2D on MI455X (gfx1250, wave32).
// GEMMs: V_WMMA_F32_16X16X4_F32, dual 16x16 tiles/wave, branchless k-loop.
// Scan: dbc rows staged to LDS via GLOBAL_LOAD_ASYNC_TO_LDS_B32, double-buffered
//       with s_wait_asynccnt, overlapping DMA with the serial recurrence.

typedef __attribute__((ext_vector_type(2))) float v2f;
typedef __attribute__((ext_vector_type(8))) float v8f;

#define B_SZ    4
#define C_MODEL 192
#define L_SEQ   4096          // H*W = 64*64
#define R_ROWS  (B_SZ * L_SEQ)
#define D_INNER 384
#define D_STATE 16
#define DT_RANK 12
#define DBC_W   44            // DT_RANK + 2*D_STATE

__device__ __forceinline__ float sigmoid_f(float x) { return 1.0f / (1.0f + expf(-x)); }

// ---------------------------------------------------------------------------
// fp32 WMMA GEMM:  Out[R x N] = act( A[R x K] * W^T + bias ), W is [N x K].
//   IMODE 0: A row-major (lda)      IMODE 1: A channel-major (B, K, L)
//   OMODE 0: Out row-major (ldo)    OMODE 1: Out channel-major (B, N, L)
//   ACT   1: softplus epilogue
// One wave32 computes a 16x32 tile (two WMMA accumulators sharing the A frag).
// Fragment layouts per CDNA5 ISA 7.12.2 (A 16x4: lanes0-15 K{0,1}, lanes16-31
// K{2,3}; B 4x16: VGPR v = row v / v+2 per lane half; C/D: VGPR v -> row
// v+8*half, col lane%16).
// ---------------------------------------------------------------------------
template <int IMODE, int OMODE, int ACT>
__global__ __launch_bounds__(256) void gemm_wmma_f32(
    const float* __restrict__ A, const float* __restrict__ W,
    const float* __restrict__ bias, float* __restrict__ Out,
    int R, int K, int N, int lda, int ldo)
{
    const int lane   = threadIdx.x & 31;
    const int wid    = (blockIdx.x * blockDim.x + threadIdx.x) >> 5;
    const int nwaves = (gridDim.x * blockDim.x) >> 5;
    const int half   = lane >> 4;
    const int l16    = lane & 15;

    const int tn    = (N + 15) >> 4;   // 16-col tiles
    const int tn2   = (tn + 1) >> 1;   // 32-col dual tiles
    const int tiles = (R >> 4) * tn2;

    for (int t = wid; t < tiles; t += nwaves) {
        const int m0  = (t / tn2) << 4;
        const int n0  = (t % tn2) << 5;
        const int nb0 = n0 + l16;
        const int nb1 = n0 + 16 + l16;
        // clamp OOB columns to a valid row: their products live only in lanes
        // that never store, so correctness is preserved with no branches.
        const int nc0 = (nb0 < N) ? nb0 : (N - 1);
        const int nc1 = (nb1 < N) ? nb1 : (N - 1);

        v8f acc0 = {0.f, 0.f, 0.f, 0.f, 0.f, 0.f, 0.f, 0.f};
        v8f acc1 = {0.f, 0.f, 0.f, 0.f, 0.f, 0.f, 0.f, 0.f};

        const int   rowA = m0 + l16;
        const float* wr0 = W + (long)nc0 * K;
        const float* wr1 = W + (long)nc1 * K;
        long abase;
        if (IMODE == 1) {
            abase = ((long)(rowA / L_SEQ) * K) * L_SEQ + (rowA % L_SEQ);
        } else {
            abase = (long)rowA * lda;
        }
        const int koff = half << 1;

#pragma unroll 2
        for (int k0 = 0; k0 < K; k0 += 4) {
            const int kk = k0 + koff;
            v2f a, b0, b1;
            if (IMODE == 1) {
                a.x = A[abase + (long)kk * L_SEQ];
                a.y = A[abase + (long)(kk + 1) * L_SEQ];
            } else {
                a = *(const v2f*)(A + abase + kk);      // 8B aligned (even strides)
            }
            b0 = *(const v2f*)(wr0 + kk);
            b1 = *(const v2f*)(wr1 + kk);
            __builtin_prefetch(wr0 + k0 + 64, 0, 0);    // global_prefetch_b8
            acc0 = __builtin_amdgcn_wmma_f32_16x16x4_f32(
                       false, a, false, b0, (short)0, acc0, false, false);
            acc1 = __builtin_amdgcn_wmma_f32_16x16x4_f32(
                       false, a, false, b1, (short)0, acc1, false, false);
        }

        // epilogue: two tiles
#pragma unroll
        for (int tile = 0; tile < 2; ++tile) {
            const int nb = tile ? nb1 : nb0;
            if (nb >= N) continue;
            const v8f& acc = tile ? acc1 : acc0;
            const float bv = bias ? bias[nb] : 0.f;
#pragma unroll
            for (int v = 0; v < 8; ++v) {
                const int r = m0 + v + (half << 3);
                float val = acc[v] + bv;
                if (ACT == 1) val = (val > 20.f) ? val : log1pf(expf(val));
                if (OMODE == 0) {
                    Out[(long)r * ldo + nb] = val;
                } else {
                    const int bb = r / L_SEQ, l = r % L_SEQ;
                    Out[((long)bb * N + nb) * L_SEQ + l] = val;
                }
            }
        }
    }
}

// ---------------------------------------------------------------------------
// LayerNorm fused with residual add: seq = LN(seq + x_chmajor) * g + b
// One wave32 per row of C_MODEL=192 (6 elements per lane).
// ---------------------------------------------------------------------------
__global__ __launch_bounds__(256) void ln_fuse_kernel(
    float* __restrict__ seq, const float* __restrict__ x,
    const float* __restrict__ gamma, const float* __restrict__ beta)
{
    const int lane = threadIdx.x & 31;
    const int row  = (blockIdx.x * blockDim.x + threadIdx.x) >> 5;
    if (row >= R_ROWS) return;
    const int b = row / L_SEQ, l = row % L_SEQ;

    float vals[6];
    float s = 0.f, s2 = 0.f;
#pragma unroll
    for (int i = 0; i < 6; ++i) {
        const int c = lane + i * 32;
        const float v = seq[(long)row * C_MODEL + c] +
                        x[((long)b * C_MODEL + c) * L_SEQ + l];
        vals[i] = v; s += v; s2 += v * v;
    }
#pragma unroll
    for (int m = 16; m >= 1; m >>= 1) {
        s  += __shfl_xor(s,  m, 32);
        s2 += __shfl_xor(s2, m, 32);
    }
    const float mu  = s * (1.0f / C_MODEL);
    const float var = s2 * (1.0f / C_MODEL) - mu * mu;
    const float inv = rsqrtf(var + 1e-5f);
#pragma unroll
    for (int i = 0; i < 6; ++i) {
        const int c = lane + i * 32;
        seq[(long)row * C_MODEL + c] = (vals[i] - mu) * inv * gamma[c] + beta[c];
    }
}

// ---------------------------------------------------------------------------
// Depthwise causal conv (k=4) along L + bias + SiLU.
// ---------------------------------------------------------------------------
__global__ __launch_bounds__(256) void conv_silu_kernel(
    const float* __restrict__ xz, const float* __restrict__ conv_w,
    const float* __restrict__ conv_b, float* __restrict__ ucv)
{
    const long tid = (long)blockIdx.x * blockDim.x + threadIdx.x;
    if (tid >= (long)R_ROWS * D_INNER) return;
    const int d   = (int)(tid % D_INNER);
    const int row = (int)(tid / D_INNER);
    const int b = row / L_SEQ, l = row % L_SEQ;

    float acc = conv_b[d];
#pragma unroll
    for (int k = 0; k < 4; ++k) {
        const int ll = l - 3 + k;
        if (ll >= 0)
            acc += xz[((long)b * L_SEQ + ll) * (2 * D_INNER) + d] * conv_w[d * 4 + k];
    }
    ucv[(long)row * D_INNER + d] = acc * sigmoid_f(acc);
}

// ---------------------------------------------------------------------------
// Selective scan. Block = 128 channels of one batch third; the (broadcast)
// dbc rows are staged into LDS in 64-row chunks with async DMA, double
// buffered so the copy of chunk c+1 overlaps the recurrence of chunk c.
// ---------------------------------------------------------------------------
#define SCAN_CHUNK  64
#define SCAN_ELEMS  (SCAN_CHUNK * DBC_W)          // 2816 used
#define SCAN_PAD    3072                          // = 24 * 128, uniform issue
#define SCAN_BLK    128
#define SCAN_ITERS  (SCAN_PAD / SCAN_BLK)         // 24 async ops / wave / chunk
#define NCHUNK      (L_SEQ / SCAN_CHUNK)          // 64

typedef __attribute__((address_space(3))) float lds_float;

__device__ __forceinline__ void async_ld_b32(unsigned lds_off, const float* gaddr)
{
    asm volatile("global_load_async_to_lds_b32 %0, %1, off"
                 :: "v"(lds_off), "v"(gaddr) : "memory");
}

__global__ __launch_bounds__(SCAN_BLK) void scan_kernel(
    const float* __restrict__ ucv, const float* __restrict__ dt,
    const float* __restrict__ dbc, const float* __restrict__ xz,
    const float* __restrict__ A_log, const float* __restrict__ Dv,
    float* __restrict__ ybuf)
{
    __shared__ float sbc[2][SCAN_PAD];

    const int tid = threadIdx.x;
    const int b   = blockIdx.x / 3;                 // batch
    const int d   = (blockIdx.x % 3) * SCAN_BLK + tid;

    const unsigned lds_base0 =
        (unsigned)(unsigned long long)(lds_float*)&sbc[0][0];
    const unsigned lds_base1 =
        (unsigned)(unsigned long long)(lds_float*)&sbc[1][0];

    float An[D_STATE], iAn[D_STATE], h[D_STATE];
#pragma unroll
    for (int n = 0; n < D_STATE; ++n) {
        const float a = -expf(A_log[d * D_STATE + n]);
        An[n] = a; iAn[n] = 1.0f / a; h[n] = 0.f;
    }
    const float Dd = Dv[d];

    // issue chunk 0
    {
        const float* src = dbc + ((long)b * L_SEQ) * DBC_W;
#pragma unroll
        for (int i = 0; i < SCAN_ITERS; ++i)
            async_ld_b32(lds_base0 + (unsigned)(i * SCAN_BLK + tid) * 4u,
                         src + i * SCAN_BLK + tid);
    }

    for (int c = 0; c < NCHUNK; ++c) {
        if (c + 1 < NCHUNK) {
            const unsigned base = ((c + 1) & 1) ? lds_base1 : lds_base0;
            const float* src = dbc + ((long)b * L_SEQ + (c + 1) * SCAN_CHUNK) * DBC_W;
#pragma unroll
            for (int i = 0; i < SCAN_ITERS; ++i)
                async_ld_b32(base + (unsigned)(i * SCAN_BLK + tid) * 4u,
                             src + i * SCAN_BLK + tid);
            asm volatile("s_wait_asynccnt 0x18" ::: "memory");  // 24 newest = next chunk
        } else {
            asm volatile("s_wait_asynccnt 0x0" ::: "memory");
        }
        __syncthreads();

        const float* bcc = &sbc[c & 1][0];
        for (int j = 0; j < SCAN_CHUNK; ++j) {
            const long row = (long)b * L_SEQ + c * SCAN_CHUNK + j;
            const float u   = ucv[row * D_INNER + d];
            const float dtv = dt [row * D_INNER + d];
            const float z   = xz [row * (2 * D_INNER) + D_INNER + d];
            const float* bc = bcc + j * DBC_W;

            float y = 0.f;
#pragma unroll
            for (int n = 0; n < D_STATE; ++n) {
                const float dA  = expf(dtv * An[n]);
                const float dBu = (dA - 1.0f) * iAn[n] * bc[DT_RANK + n] * u;
                h[n] = dA * h[n] + dBu;
                y += h[n] * bc[DT_RANK + D_STATE + n];
            }
            ybuf[row * D_INNER + d] = (y + u * Dd) * (z * sigmoid_f(z));
        }
        __syncthreads();
    }
}

// ---------------------------------------------------------------------------
extern "C" void kernel_launch(void* const* d_in, const int* in_sizes, int n_in,
                              void* d_out, int out_size, void* d_ws, size_t ws_size,
                              hipStream_t stream)
{
    const float* x        = (const float*)d_in[0];
    const float* guide    = (const float*)d_in[1];
    const float* guide_w  = (const float*)d_in[2];
    const float* ln_g     = (const float*)d_in[3];
    const float* ln_b     = (const float*)d_in[4];
    const float* in_projw = (const float*)d_in[5];   // (768, 192)
    const float* conv_w   = (const float*)d_in[6];   // (384, 4)
    const float* conv_b   = (const float*)d_in[7];
    const float* x_projw  = (const float*)d_in[8];   // (44, 384)
    const float* dt_projw = (const float*)d_in[9];   // (384, 12)
    const float* dt_projb = (const float*)d_in[10];
    const float* A_log    = (const float*)d_in[11];  // (384, 16)
    const float* Dv       = (const float*)d_in[12];
    const float* out_projw= (const float*)d_in[13];  // (192, 384)
    float* out = (float*)d_out;

    // workspace carve (floats)
    float* p = (float*)d_ws;
    float* seqbuf = p; p += (long)R_ROWS * C_MODEL;       // (R,192)
    float* xz     = p; p += (long)R_ROWS * 2 * D_INNER;   // (R,768)
    float* ucv    = p; p += (long)R_ROWS * D_INNER;       // (R,384)
    float* dbc    = p; p += (long)R_ROWS * 48;            // (R,44) padded
    float* dtb    = p; p += (long)R_ROWS * D_INNER;       // (R,384)
    float* ybuf   = p; p += (long)R_ROWS * D_INNER;       // (R,384)

    const dim3 gblk(2048), gthr(256);

    // 1) guide channel-mix: seqbuf = guide_w * guide   (A channel-major)
    gemm_wmma_f32<1, 0, 0><<<gblk, gthr, 0, stream>>>(
        guide, guide_w, nullptr, seqbuf, R_ROWS, C_MODEL, C_MODEL, 0, C_MODEL);
    // 2) seqbuf = LN(seqbuf + x) * g + b
    ln_fuse_kernel<<<R_ROWS / 8, 256, 0, stream>>>(seqbuf, x, ln_g, ln_b);
    // 3) xz = seqbuf @ in_proj_w^T  (R,768)
    gemm_wmma_f32<0, 0, 0><<<gblk, gthr, 0, stream>>>(
        seqbuf, in_projw, nullptr, xz, R_ROWS, C_MODEL, 2 * D_INNER, C_MODEL, 2 * D_INNER);
    // 4) ucv = silu(conv(xz[:, :384]) + conv_b)
    {
        long n = (long)R_ROWS * D_INNER;
        conv_silu_kernel<<<(unsigned)((n + 255) / 256), 256, 0, stream>>>(
            xz, conv_w, conv_b, ucv);
    }
    // 5) dbc = ucv @ x_proj_w^T  (R,44)
    gemm_wmma_f32<0, 0, 0><<<gblk, gthr, 0, stream>>>(
        ucv, x_projw, nullptr, dbc, R_ROWS, D_INNER, DBC_W, D_INNER, DBC_W);
    // 6) dt = softplus(dbc[:, :12] @ dt_proj_w^T + dt_proj_b)  (R,384)
    gemm_wmma_f32<0, 0, 1><<<gblk, gthr, 0, stream>>>(
        dbc, dt_projw, dt_projb, dtb, R_ROWS, DT_RANK, D_INNER, DBC_W, D_INNER);
    // 7) selective scan + gating -> ybuf  (async dbc staging to LDS)
    scan_kernel<<<B_SZ * 3, SCAN_BLK, 0, stream>>>(
        ucv, dtb, dbc, xz, A_log, Dv, ybuf);
    // 8) out = ybuf @ out_proj_w^T, stored channel-major (B, 192, H, W)
    gemm_wmma_f32<0, 1, 0><<<gblk, gthr, 0, stream>>>(
        ybuf, out_projw, nullptr, out, R_ROWS, D_INNER, C_MODEL, D_INNER, 0);
}